// CTC_90417651516567
// MI455X (gfx1250) — compile-verified
//
#include <hip/hip_runtime.h>
#include <hip/hip_bf16.h>

typedef float v2f __attribute__((ext_vector_type(2)));
typedef float v8f __attribute__((ext_vector_type(8)));

#define B_DIM   32
#define T_DIM   2048
#define D_DIM   512
#define V_DIM   5
#define L_DIM   256
#define S_EXT   (2 * L_DIM + 1)      // 513 extended CTC states

#define WAVES_PER_WG   8
#define ROWS_PER_WAVE  16
#define ROWS_PER_WG    (WAVES_PER_WG * ROWS_PER_WAVE)   // 128
#define KSTEPS         (D_DIM / 4)                       // 128 wmma steps of K=4

// ---------------------------------------------------------------------------
// Kernel 1: logits = hs @ W + b  (skinny GEMM via V_WMMA_F32_16X16X4_F32,
// V padded 5->16), fused log-softmax over V, writes log-probs and probs.
// ---------------------------------------------------------------------------
__global__ __launch_bounds__(256) void ctc_gemm_logsoftmax_kernel(
    const float* __restrict__ hs,     // (B*T, 512) row-major
    const float* __restrict__ W,      // (512, 5)
    const float* __restrict__ bias,   // (5)
    float* __restrict__ out_lp,       // (B*T, 5)  log-probs  (== ys_hat)
    float* __restrict__ out_p,        // (B*T, 5)  probs
    float* __restrict__ loss_out)     // scalar, zero-init here
{
    // B fragments pre-swizzled into exact WMMA-B lane layout:
    //   Wsw[j*32 + l] = { Wp[4j + 2*(l>>4)][l&15], Wp[4j + 2*(l>>4) + 1][l&15] }
    __shared__ v2f   Wsw[KSTEPS * 32];            // 32 KB
    __shared__ float logits[ROWS_PER_WG][8];      // 4 KB (cols 0..4 used)

    const int tid = threadIdx.x;
    if (blockIdx.x == 0 && tid == 0) *loss_out = 0.0f;

    for (int idx = tid; idx < KSTEPS * 32; idx += blockDim.x) {
        const int j = idx >> 5;
        const int l = idx & 31;
        const int k = 4 * j + 2 * (l >> 4);
        const int n = l & 15;
        v2f w;
        w.x = (n < V_DIM) ? W[k * V_DIM + n] : 0.0f;
        w.y = (n < V_DIM) ? W[(k + 1) * V_DIM + n] : 0.0f;
        Wsw[idx] = w;
    }
    __syncthreads();

    const int wave = tid >> 5;
    const int lane = tid & 31;
    const int m    = lane & 15;            // tile row (A) / column (C)
    const int half = lane >> 4;            // 0: C rows 0-7, 1: C rows 8-15
    const int koff = half << 1;            // A K sub-offset {0,2}

    const long rowBase = (long)blockIdx.x * ROWS_PER_WG + (long)wave * ROWS_PER_WAVE;
    const float* aPtr  = hs + (rowBase + m) * D_DIM + koff;

    v8f acc = {0.f, 0.f, 0.f, 0.f, 0.f, 0.f, 0.f, 0.f};
    #pragma unroll 1
    for (int j2 = 0; j2 < KSTEPS / 8; ++j2) {
        __builtin_prefetch(aPtr + 32 * (j2 + 2), 0, 3);   // global_prefetch_b8
        #pragma unroll
        for (int u = 0; u < 8; ++u) {
            const int j = j2 * 8 + u;
            v2f a   = *(const v2f*)(aPtr + 4 * j);
            v2f bfr = Wsw[j * 32 + lane];
            acc = __builtin_amdgcn_wmma_f32_16x16x4_f32(
                false, a, false, bfr, (short)0, acc, false, false);
        }
    }

    // Spill valid columns (N < 5) of the 16x16 accumulator to LDS.
    // acc[v] in lane l holds C[row = v + 8*(l>>4)][col = l&15].
    if (m < V_DIM) {
        #pragma unroll
        for (int v = 0; v < 8; ++v)
            logits[wave * 16 + v + 8 * half][m] = acc[v];
    }
    __syncthreads();

    // Per-row log-softmax over V=5; one thread per row.
    if (tid < ROWS_PER_WG) {
        const long R = (long)blockIdx.x * ROWS_PER_WG + tid;
        float x[V_DIM];
        float mx = -3.4e38f;
        #pragma unroll
        for (int v = 0; v < V_DIM; ++v) {
            x[v] = logits[tid][v] + bias[v];
            mx = fmaxf(mx, x[v]);
        }
        float se = 0.0f;
        #pragma unroll
        for (int v = 0; v < V_DIM; ++v) se += __expf(x[v] - mx);
        const float lse = mx + __logf(se);
        #pragma unroll
        for (int v = 0; v < V_DIM; ++v) {
            const float lp = x[v] - lse;
            out_lp[R * V_DIM + v] = lp;
            out_p [R * V_DIM + v] = __expf(lp);
        }
    }
}

// ---------------------------------------------------------------------------
// Kernel 2: CTC forward (alpha recursion) in log space.
// One workgroup per sample; one thread per extended state; alpha double-
// buffered in LDS -> exactly one barrier per timestep.
// ---------------------------------------------------------------------------
__global__ __launch_bounds__(544) void ctc_forward_kernel(
    const float* __restrict__ lp,      // (B, T, 5) log-probs
    const int*   __restrict__ labels,  // (B, 256)
    const int*   __restrict__ hlens,   // (B,)
    const int*   __restrict__ yslens,  // (B,)
    float* __restrict__ loss_out)
{
    __shared__ float abuf[2][S_EXT + 3];

    const int   b   = blockIdx.x;
    const int   s   = threadIdx.x;
    const float NEG = -1e9f;
    const float* lpb = lp + (long)b * T_DIM * V_DIM;

    int  ext_s = 0;      // extended symbol for state s (blank at even s)
    bool skip  = false;  // s-2 -> s transition allowed
    if (s < S_EXT && (s & 1)) {
        ext_s = labels[b * L_DIM + ((s - 1) >> 1)];
        if (s >= 3) skip = (ext_s != labels[b * L_DIM + ((s - 3) >> 1)]);
    }

    if (s < S_EXT) {
        float a0 = NEG;
        if (s == 0)      a0 = lpb[0];           // blank
        else if (s == 1) a0 = lpb[ext_s];       // first label
        abuf[0][s] = a0;
    }
    __syncthreads();

    const int hl = hlens[b];
    int cur = 0;
    for (int t = 1; t < T_DIM; ++t) {
        float nv = NEG;
        float a  = NEG;
        if (s < S_EXT) {
            const float lp_my = lpb[t * V_DIM + ext_s];
            a = abuf[cur][s];
            const float a1 = (s >= 1) ? abuf[cur][s - 1] : NEG;
            const float a2 = skip     ? abuf[cur][s - 2] : NEG;
            const float mm = fmaxf(a, fmaxf(a1, a2));
            float r;
            if (mm <= -0.5e9f) r = NEG;
            else r = mm + __logf(__expf(a - mm) + __expf(a1 - mm) + __expf(a2 - mm));
            nv = r + lp_my;
            if (t >= hl) nv = a;                // past input length: freeze alpha
            abuf[cur ^ 1][s] = nv;
        }
        cur ^= 1;
        __syncthreads();
    }

    if (s == 0) {
        const int i1 = 2 * yslens[b] - 1;
        const float aL = abuf[cur][i1];
        const float aB = abuf[cur][i1 + 1];
        const float mm = fmaxf(aL, aB);
        float nll;
        if (mm <= -0.5e9f) nll = 1e30f;
        else nll = -(mm + __logf(__expf(aL - mm) + __expf(aB - mm)));
        if (nll > 1e8f) nll = 0.0f;             // zero_infinity=True
        atomicAdd(loss_out, nll / (float)B_DIM);
    }
}

// ---------------------------------------------------------------------------
extern "C" void kernel_launch(void* const* d_in, const int* in_sizes, int n_in,
                              void* d_out, int out_size, void* d_ws, size_t ws_size,
                              hipStream_t stream) {
    const float* hs     = (const float*)d_in[0];   // (B, T, D)
    const float* W      = (const float*)d_in[1];   // (D, V)
    const float* bias   = (const float*)d_in[2];   // (V,)
    const int*   hlens  = (const int*)  d_in[3];   // (B,)
    const int*   ys_pad = (const int*)  d_in[4];   // (B, L)
    const int*   ys_len = (const int*)  d_in[5];   // (B,)

    float* out    = (float*)d_out;
    float* loss   = out;                                   // [0]
    float* ys_hat = out + 1;                               // (B, T, V) log-probs
    float* probs  = out + 1 + (long)B_DIM * T_DIM * V_DIM; // (B, T, V) probs

    const int n_rows = B_DIM * T_DIM;                      // 65536
    ctc_gemm_logsoftmax_kernel<<<n_rows / ROWS_PER_WG, 256, 0, stream>>>(
        hs, W, bias, ys_hat, probs, loss);

    ctc_forward_kernel<<<B_DIM, 544, 0, stream>>>(
        ys_hat, ys_pad, hlens, ys_len, loss);
}